// RGCNEncoder_83708912599118
// MI455X (gfx1250) — compile-verified
//
#include <hip/hip_runtime.h>
#include <stdint.h>

#define NN 100000   // nodes
#define RR 16       // relations
#define EE 50000    // edges per relation
#define DD 500      // feature dim
#define LDB 512     // padded B leading dim
#define KC 100      // K-chunk staged in LDS per TDM transfer (500 = 5*KC)

typedef float v2f __attribute__((ext_vector_type(2)));
typedef float v8f __attribute__((ext_vector_type(8)));
typedef unsigned v4u __attribute__((ext_vector_type(4)));
typedef int v4i __attribute__((ext_vector_type(4)));
typedef int v8i __attribute__((ext_vector_type(8)));

__device__ __forceinline__ void fatomic_add(float* p, float v) {
  __hip_atomic_fetch_add(p, v, __ATOMIC_RELAXED, __HIP_MEMORY_SCOPE_AGENT);
}

__device__ __forceinline__ unsigned rotl32(unsigned x, int n) {
  return (x << n) | (x >> (32 - n));
}

// ---------------------------------------------------------------------------
// Edge dropout mask (exact JAX threefry2x32, key = (0,42)) + per-relation
// in-degree counts. Computed once; shared by both layers.
// ---------------------------------------------------------------------------
__global__ __launch_bounds__(256) void mask_count_kernel(
    const unsigned char* __restrict__ emask, const int* __restrict__ idcs,
    float* __restrict__ maskf, unsigned* __restrict__ cnt)
{
  int i = blockIdx.x * 256 + threadIdx.x;
  const int total = RR * EE;          // 800000 (even)
  if (i >= total) return;
  const unsigned half = (unsigned)(total >> 1);
  unsigned j = ((unsigned)i < half) ? (unsigned)i : (unsigned)i - half;
  unsigned x0 = j, x1 = j + half;
  const unsigned ks0 = 0u, ks1 = 42u, ks2 = 0x1BD11BDAu ^ ks0 ^ ks1;
  x0 += ks0; x1 += ks1;
#define TF_ROUND(rot) { x0 += x1; x1 = rotl32(x1, rot); x1 ^= x0; }
  TF_ROUND(13) TF_ROUND(15) TF_ROUND(26) TF_ROUND(6)
  x0 += ks1; x1 += ks2 + 1u;
  TF_ROUND(17) TF_ROUND(29) TF_ROUND(16) TF_ROUND(24)
  x0 += ks2; x1 += ks0 + 2u;
  TF_ROUND(13) TF_ROUND(15) TF_ROUND(26) TF_ROUND(6)
  x0 += ks0; x1 += ks1 + 3u;
  TF_ROUND(17) TF_ROUND(29) TF_ROUND(16) TF_ROUND(24)
  x0 += ks1; x1 += ks2 + 4u;
  TF_ROUND(13) TF_ROUND(15) TF_ROUND(26) TF_ROUND(6)
  x0 += ks2; x1 += ks0 + 5u;
#undef TF_ROUND
  unsigned bits = ((unsigned)i < half) ? x0 : x1;
  float u = __uint_as_float((bits >> 9) | 0x3f800000u) - 1.0f;   // [0,1)
  float m = (emask[i] && (u < 0.5f)) ? 1.0f : 0.0f;
  maskf[i] = m;
  if (m != 0.0f) {
    int r = i / EE, e = i - r * EE;
    int d = idcs[(size_t)r * 2 * EE + EE + e];   // dst index
    __hip_atomic_fetch_add(&cnt[(size_t)r * NN + d], 1u,
                           __ATOMIC_RELAXED, __HIP_MEMORY_SCOPE_AGENT);
  }
}

// ---------------------------------------------------------------------------
// Zero-pad w_self [500x500] -> Bpad [500x512] so GEMM B tiles are unguarded.
// ---------------------------------------------------------------------------
__global__ __launch_bounds__(256) void padB_kernel(
    const float* __restrict__ w, float* __restrict__ Bp)
{
  int i = blockIdx.x * 256 + threadIdx.x;   // 500*512 elements
  if (i >= DD * LDB) return;
  int k = i / LDB, n = i - k * LDB;
  Bp[i] = (n < DD) ? w[k * DD + n] : 0.0f;
}

// ---------------------------------------------------------------------------
// C[M,500] = A[M,500] @ Bpad[500,512] with V_WMMA_F32_16X16X4_F32.
// Block = 256 threads = 8 waves stacked in M; each wave: 32(M) x 64(N) tile.
// B panel is streamed into LDS by the Tensor Data Mover (tensor_load_to_lds),
// double-buffered in K-chunks of 100, synchronized with s_wait_tensorcnt +
// workgroup barrier; DMA of chunk c+1 overlaps WMMA compute on chunk c.
//
// Fragments:
//  A (16x4 f32): lane l (m=l%16, hi=l/16) holds K = 2hi, 2hi+1 in 2 VGPRs.
//  B (4x16 f32): lane l (n=l%16, hi=l/16) holds K = 2hi, 2hi+1.
//  C/D (16x16 f32): element (m,n) at vgpr (m%8), lane = (m/8)*16 + n.
// ---------------------------------------------------------------------------
__device__ __forceinline__ v8f wmma4(v2f a, v2f b, v8f c) {
  return __builtin_amdgcn_wmma_f32_16x16x4_f32(
      false, a, false, b, (short)0, c, false, false);
}

__global__ __launch_bounds__(256) void gemm_wmma_kernel(
    const float* __restrict__ A, const float* __restrict__ Bp,
    float* __restrict__ C)
{
  const int lane = threadIdx.x & 31;
  const int wave = threadIdx.x >> 5;
  int m0 = blockIdx.x * 256 + wave * 32;
  const bool mvalid = (m0 < NN);              // NN % 32 == 0 -> full tiles only
  if (!mvalid) m0 = 0;                        // clamp loads; stores suppressed
  const int n0 = blockIdx.y * 64;
  const int mr = lane & 15;
  const int hi = lane >> 4;

  __shared__ float lb[2][KC * 64];            // 2 x 25.6 KB TDM landing pads

  // TDM descriptor: 2-D tile (64 cols x KC rows of f32) out of Bpad[500][512],
  // packed row-major into LDS. D# group bitfields per CDNA5 ISA ch.8.
  auto issue_tdm = [&](int chunk, int bufsel) {
    unsigned lds_addr = (unsigned)(size_t)(&lb[bufsel][0]);
    unsigned long long ga =
        (unsigned long long)(size_t)(Bp + (size_t)chunk * KC * LDB + n0);
    v4u g0;
    g0[0] = 1u;                               // count=1, no gather/iterate
    g0[1] = lds_addr;                         // lds_addr [63:32]
    g0[2] = (unsigned)(ga & 0xffffffffu);     // global_addr lo
    g0[3] = (unsigned)(ga >> 32) | (2u << 30);// global_addr hi | type=2
    v8i g1;
    g1[0] = (int)(2u << 16);                  // wg_mask=0 | data_size=4B
    g1[1] = (int)((unsigned)LDB << 16);       // tensor_dim0[15:0]=512
    g1[2] = (int)((unsigned)DD << 16);        // dim0 hi=0 | tensor_dim1 lo=500
    g1[3] = (int)(64u << 16);                 // dim1 hi=0 | tile_dim0=64
    g1[4] = (int)KC;                          // tile_dim1=100 | tile_dim2=0
    g1[5] = (int)LDB;                         // tensor_dim0_stride=512
    g1[6] = 0;                                // stride0 hi | stride1 lo
    g1[7] = 0;                                // stride1 hi
    v4i gz4 = {0, 0, 0, 0};                   // 2-D tensor: groups 2/3 unused
    v8i gz8 = {0, 0, 0, 0, 0, 0, 0, 0};
    __builtin_amdgcn_tensor_load_to_lds(g0, g1, gz4, gz4, gz8, 0);
  };

  v8f acc0[4] = {};
  v8f acc1[4] = {};

  if (wave == 0) issue_tdm(0, 0);

  for (int c = 0; c < DD / KC; ++c) {
    __builtin_amdgcn_s_wait_tensorcnt(0);     // chunk c resident (wave0; others 0)
    __syncthreads();                          // publish LDS; prev compute done
    if (wave == 0 && c + 1 < DD / KC)
      issue_tdm(c + 1, (c + 1) & 1);          // overlap DMA with compute

    const float* A0 = A + (size_t)(m0 + mr) * DD + c * KC;
    const float* A1 = A0 + (size_t)16 * DD;
    const float* Lb = &lb[c & 1][mr];
#pragma unroll 1
    for (int kk = 0; kk < KC; kk += 4) {
      const int ka = kk + 2 * hi;
      v2f a0 = *(const v2f*)(A0 + ka);        // 8B aligned (ka even)
      v2f a1 = *(const v2f*)(A1 + ka);
      const float* Lr = Lb + ka * 64;
#pragma unroll
      for (int t = 0; t < 4; ++t) {
        v2f b;
        b.x = Lr[t * 16];
        b.y = Lr[t * 16 + 64];
        acc0[t] = wmma4(a0, b, acc0[t]);
        acc1[t] = wmma4(a1, b, acc1[t]);
      }
    }
  }

  if (!mvalid) return;
#pragma unroll
  for (int t = 0; t < 4; ++t) {
    const int n = n0 + t * 16 + mr;
    if (n >= DD) continue;                    // trim padded columns
#pragma unroll
    for (int v = 0; v < 8; ++v) {
      C[(size_t)(m0 + 8 * hi + v) * DD + n]      = acc0[t][v];
      C[(size_t)(m0 + 16 + 8 * hi + v) * DD + n] = acc1[t][v];
    }
  }
}

// ---------------------------------------------------------------------------
// Per-edge block-diagonal matmul + normalized scatter-add:
//   out[dst] += (h[src] blockdot w_rel[r]) * mask / max(cnt[r][dst],1)
// One wave per edge; w_rel[r] (10KB) staged in LDS; dropped edges skipped.
// ---------------------------------------------------------------------------
#define EPW 32  // edges per wave
__global__ __launch_bounds__(256) void scatter_kernel(
    const float* __restrict__ h, const float* __restrict__ w_rel,
    const int* __restrict__ idcs, const float* __restrict__ maskf,
    const unsigned* __restrict__ cnt, float* __restrict__ out)
{
  const int r = blockIdx.y;
  const int wave = threadIdx.x >> 5;
  const int lane = threadIdx.x & 31;

  __shared__ float ws[2500];                  // w_rel[r]: 100 blocks x 5 x 5
  for (int i = threadIdx.x; i < 2500; i += 256)
    ws[i] = w_rel[(size_t)r * 2500 + i];
  __syncthreads();

  const int* src = idcs + (size_t)r * 2 * EE;
  const int* dst = src + EE;
  const float* mrow = maskf + (size_t)r * EE;
  const unsigned* crow = cnt + (size_t)r * NN;

  const int e0 = blockIdx.x * 256 + wave * EPW;
#pragma unroll 1
  for (int t = 0; t < EPW; ++t) {
    const int e = e0 + t;
    if (e >= EE) break;
    const float m = mrow[e];
    if (m == 0.0f) continue;                  // dropped edge: zero contribution
    const int s = src[e];
    const int d = dst[e];
    const unsigned c = crow[d];
    const float scale = m / (float)(c > 1u ? c : 1u);
    const float* hrow = h + (size_t)s * DD;
    float* orow = out + (size_t)d * DD;
#pragma unroll 1
    for (int b = lane; b < 100; b += 32) {    // lane-parallel over 5x5 blocks
      const float* wb = ws + b * 25;
      const float* hx = hrow + b * 5;
      const float x0 = hx[0], x1 = hx[1], x2 = hx[2], x3 = hx[3], x4 = hx[4];
#pragma unroll
      for (int o = 0; o < 5; ++o) {
        float v = x0 * wb[o] + x1 * wb[5 + o] + x2 * wb[10 + o]
                + x3 * wb[15 + o] + x4 * wb[20 + o];
        fatomic_add(&orow[b * 5 + o], v * scale);
      }
    }
  }
}

// ---------------------------------------------------------------------------
__global__ __launch_bounds__(256) void relu_kernel(float* __restrict__ p, long n4)
{
  long i = (long)blockIdx.x * 256 + threadIdx.x;
  if (i >= n4) return;
  float4 v = ((float4*)p)[i];
  v.x = fmaxf(v.x, 0.0f); v.y = fmaxf(v.y, 0.0f);
  v.z = fmaxf(v.z, 0.0f); v.w = fmaxf(v.w, 0.0f);
  ((float4*)p)[i] = v;
}

// ---------------------------------------------------------------------------
extern "C" void kernel_launch(void* const* d_in, const int* in_sizes, int n_in,
                              void* d_out, int out_size, void* d_ws, size_t ws_size,
                              hipStream_t stream)
{
  const float* x       = (const float*)d_in[0];
  const float* w_rel1  = (const float*)d_in[1];
  const float* w_self1 = (const float*)d_in[2];
  const float* w_rel2  = (const float*)d_in[3];
  const float* w_self2 = (const float*)d_in[4];
  const int*   idcs    = (const int*)d_in[5];
  const unsigned char* emask = (const unsigned char*)d_in[6];
  float* out = (float*)d_out;

  char* p = (char*)d_ws;
  auto take = [&](size_t bytes) {
    void* q = (void*)p;
    p += (bytes + 255) & ~(size_t)255;
    return q;
  };
  float*    maskf = (float*)   take((size_t)RR * EE * sizeof(float));     // 3.2 MB
  unsigned* cnt   = (unsigned*)take((size_t)RR * NN * sizeof(unsigned));  // 6.4 MB
  float*    Bpad  = (float*)   take((size_t)DD * LDB * sizeof(float));    // 1 MB
  float*    h1    = (float*)   take((size_t)NN * DD * sizeof(float));     // 200 MB

  (void)hipMemsetAsync(cnt, 0, (size_t)RR * NN * sizeof(unsigned), stream);
  mask_count_kernel<<<(RR * EE + 255) / 256, 256, 0, stream>>>(emask, idcs, maskf, cnt);

  dim3 ggrid((NN + 255) / 256, LDB / 64);
  dim3 sgrid((EE + 255) / 256, RR);
  const long n4 = (long)NN * DD / 4;
  const int relu_blocks = (int)((n4 + 255) / 256);

  // ---- layer 1: h1 = relu(x @ w_self1 + messages(x)) ----
  padB_kernel<<<(DD * LDB + 255) / 256, 256, 0, stream>>>(w_self1, Bpad);
  gemm_wmma_kernel<<<ggrid, 256, 0, stream>>>(x, Bpad, h1);
  scatter_kernel<<<sgrid, 256, 0, stream>>>(x, w_rel1, idcs, maskf, cnt, h1);
  relu_kernel<<<relu_blocks, 256, 0, stream>>>(h1, n4);

  // ---- layer 2: out = relu(h1 @ w_self2 + messages(h1)) ----
  padB_kernel<<<(DD * LDB + 255) / 256, 256, 0, stream>>>(w_self2, Bpad);
  gemm_wmma_kernel<<<ggrid, 256, 0, stream>>>(h1, Bpad, out);
  scatter_kernel<<<sgrid, 256, 0, stream>>>(h1, w_rel2, idcs, maskf, cnt, out);
  relu_kernel<<<relu_blocks, 256, 0, stream>>>(out, n4);
}